// KVCache_8512625181195
// MI455X (gfx1250) — compile-verified
//
#include <hip/hip_runtime.h>
#include <math.h>

// ---------------------------------------------------------------------------
// KV-cache shift + RoPE for MI455X (gfx1250).
// Pure streaming problem: ~512 MB of traffic, ~22 us at 23.3 TB/s HBM.
// No matmul -> no WMMA. Optimize the data-movement path instead:
//   * b128 (v4f) coalesced, non-temporal loads/stores
//   * async global<->LDS copies (ASYNCcnt) for the pure-copy V half
//   * sin/cos computed once per (position, freq) and reused across 16 heads
// ---------------------------------------------------------------------------

typedef float v4f __attribute__((ext_vector_type(4)));
typedef int   v4i __attribute__((vector_size(4 * sizeof(int))));  // builtin's pointee type

#define AS1 __attribute__((address_space(1)))
#define AS3 __attribute__((address_space(3)))

// Problem constants (fixed by the reference).
constexpr int B        = 4;      // batch
constexpr int PPS      = 256;    // pages per sequence (KV_LEN / PAGE_SIZE)
constexpr int PAGE     = 16;     // positions per page
constexpr int H        = 16;     // heads
constexpr int D        = 128;    // head dim
constexpr int SEQ      = 512;    // new tokens per batch
constexpr int HALF     = PAGE * H * D;   // 32768 floats: one kv-half of a page
constexpr int PELEM    = 2 * HALF;       // 65536 floats per page
constexpr int INPUT_P0 = 223;    // first page sourced from k/v inputs (3568/16)
constexpr int SHIFT_PG = 33;     // shift is exactly 33 pages (528/16)

// gfx12-style cache policy immediate: TH[2:0]=1 -> non-temporal, scope=default.
constexpr int CPOL_NT = 1;

// log2(10000)/64, so inv_freq[i] = 2^(-i * LOG2C)
constexpr double LOG2C = 0.20762050593046010;

#if __has_builtin(__builtin_amdgcn_global_load_async_to_lds_b128) && \
    __has_builtin(__builtin_amdgcn_global_store_async_from_lds_b128)
#define USE_ASYNC_LDS 1
#else
#define USE_ASYNC_LDS 0
#endif

// Uniform (per-block) source pointer for logical page p of batch b.
// half_off selects the K (0) or V (HALF) half of a cache page; kvsrc is k or v.
__device__ __forceinline__ const float* page_src(int b, int p, int half_off,
                                                 const float* __restrict__ kvsrc,
                                                 const float* __restrict__ cache,
                                                 const int* __restrict__ idx) {
  if (p < INPUT_P0)
    return cache + (size_t)idx[b * PPS + p + SHIFT_PG] * PELEM + half_off;
  if (p < PPS - 1)
    return kvsrc + (size_t)(b * SEQ + (p - INPUT_P0) * PAGE) * (H * D);
  return cache + (size_t)idx[b * PPS + p] * PELEM + half_off;   // tail page: self
}

// ---------------------------------------------------------------------------
// K half: gather page block + rotate-half RoPE at position j. One block per
// (b, page): 256 threads = 16 positions x 16 dim-quads, head loop inside so
// each thread's 4 sincos are reused 16x.
// ---------------------------------------------------------------------------
__global__ void __launch_bounds__(256)
rope_k_kernel(const float* __restrict__ k, const float* __restrict__ cache,
              const int* __restrict__ idx, float* __restrict__ out) {
  const int blk = blockIdx.x;
  const int b = blk / PPS, p = blk - b * PPS;

  const float* src = page_src(b, p, 0, k, cache, idx);
  float* dst = out + (size_t)idx[b * PPS + p] * PELEM;

  const int tid = threadIdx.x;
  const int pos = tid >> 4;        // 0..15 position in page
  const int q   = tid & 15;        // dim-quad: freqs i = 4q..4q+3
  const int j   = p * PAGE + pos;  // absolute position -> RoPE angle

  float c[4], s[4];
#pragma unroll
  for (int t = 0; t < 4; ++t) {
    const int i = q * 4 + t;
    const float inv = (float)exp2(-(double)i * LOG2C);   // 10000^(-i/64)
    const float a = (float)((double)j * (double)inv);
    c[t] = cosf(a);
    s[t] = sinf(a);
  }

  const v4f* s4 = (const v4f*)src;
  v4f* d4 = (v4f*)dst;
#pragma unroll 8
  for (int h = 0; h < H; ++h) {
    const int f1 = pos * (H * D / 4) + h * (D / 4) + q;  // dims 4q..4q+3
    const int f2 = f1 + 16;                              // dims +64
    v4f x1 = __builtin_nontemporal_load(s4 + f1);
    v4f x2 = __builtin_nontemporal_load(s4 + f2);
    v4f o1, o2;
#pragma unroll
    for (int t = 0; t < 4; ++t) {
      o1[t] = x1[t] * c[t] - x2[t] * s[t];
      o2[t] = x1[t] * s[t] + x2[t] * c[t];
    }
    __builtin_nontemporal_store(o1, d4 + f1);
    __builtin_nontemporal_store(o2, d4 + f2);
  }
}

// ---------------------------------------------------------------------------
// V half: pure 128KB-contiguous block gather/scatter. 4 blocks per (b, page),
// each block moves a 32KB chunk. Async global->LDS->global path (ASYNCcnt)
// when the gfx1250 async builtins exist, else plain non-temporal v4f copy.
// Each wave stores exactly the LDS regions it loaded itself (tid-indexed),
// so s_wait_asynccnt 0 between phases is sufficient -- no barrier needed.
// ---------------------------------------------------------------------------
constexpr int VSPLIT  = 4;
constexpr int F4_HALF = HALF / 4;            // 8192 v4f per page-half
constexpr int F4_BLK  = F4_HALF / VSPLIT;    // 2048 v4f per block (32 KB)
constexpr int VITERS  = F4_BLK / 256;        // 8 v4f per thread

__global__ void __launch_bounds__(256)
copy_v_kernel(const float* __restrict__ v, const float* __restrict__ cache,
              const int* __restrict__ idx, float* __restrict__ out) {
  const int blk = blockIdx.x;
  const int split = blk & (VSPLIT - 1);
  const int bp = blk >> 2;
  const int b = bp / PPS, p = bp - b * PPS;

  const float* src = page_src(b, p, HALF, v, cache, idx);
  float* dst = out + (size_t)idx[b * PPS + p] * PELEM + HALF;

  const v4f* s4 = (const v4f*)src + split * F4_BLK;
  v4f* d4 = (v4f*)dst + split * F4_BLK;
  const int tid = threadIdx.x;

#if USE_ASYNC_LDS
  __shared__ v4f lbuf[F4_BLK];                       // 32 KB staging
  AS3 v4i* lds = (AS3 v4i*)lbuf;
#pragma unroll
  for (int i = 0; i < VITERS; ++i)
    __builtin_amdgcn_global_load_async_to_lds_b128(
        (AS1 v4i*)(s4 + tid + i * 256),
        lds + tid + i * 256, 0, CPOL_NT);
  asm volatile("s_wait_asynccnt 0" ::: "memory");     // loads landed in LDS
#pragma unroll
  for (int i = 0; i < VITERS; ++i)
    __builtin_amdgcn_global_store_async_from_lds_b128(
        (AS1 v4i*)(d4 + tid + i * 256),
        lds + tid + i * 256, 0, CPOL_NT);
  asm volatile("s_wait_asynccnt 0" ::: "memory");     // stores drained
#else
#pragma unroll
  for (int i = 0; i < VITERS; ++i) {
    v4f x = __builtin_nontemporal_load(s4 + tid + i * 256);
    __builtin_nontemporal_store(x, d4 + tid + i * 256);
  }
#endif
}

// ---------------------------------------------------------------------------
extern "C" void kernel_launch(void* const* d_in, const int* in_sizes, int n_in,
                              void* d_out, int out_size, void* d_ws, size_t ws_size,
                              hipStream_t stream) {
  // setup_inputs order: k, v, draft_cache, kv_page_indices, scalars...
  const float* k     = (const float*)d_in[0];
  const float* v     = (const float*)d_in[1];
  const float* cache = (const float*)d_in[2];
  const int*   idx   = (const int*)d_in[3];
  float* out = (float*)d_out;

  rope_k_kernel<<<B * PPS, 256, 0, stream>>>(k, cache, idx, out);
  copy_v_kernel<<<B * PPS * VSPLIT, 256, 0, stream>>>(v, cache, idx, out);
}